// StableStateSSM_16999480558026
// MI455X (gfx1250) — compile-verified
//
#include <hip/hip_runtime.h>
#include <hip/hip_bf16.h>

// CDNA5 (gfx1250) StableStateSSM.
// Phase GEMMs use V_WMMA_F32_16X16X4_F32 (fp32 in == fp32 reference precision;
// op is HBM-bandwidth-bound at ~6 FLOP/byte so fp32 WMMA is the right ceiling).
// The nonlinear scan is latency-bound: 2 waves, state in registers, cross-lane
// broadcasts (ds_bpermute) + width-16 shfl_xor reduction trees.

typedef __attribute__((ext_vector_type(2))) float v2f;
typedef __attribute__((ext_vector_type(8))) float v8f;

#define BDIM   4
#define LDIM   4096
#define DDIM   1024
#define SDIM   16
#define NTOK   (BDIM * LDIM)      // 16384 tokens
#define NTILE  (NTOK / 16)        // 1024 16-token tiles

__device__ __forceinline__ v8f wmma4(v2f a, v2f b, v8f c) {
  // 8-arg f32 pattern: (neg_a, A, neg_b, B, c_mod, C, reuse_a, reuse_b)
  return __builtin_amdgcn_wmma_f32_16x16x4_f32(false, a, false, b, (short)0, c,
                                               false, false);
}

// ---------------------------------------------------------------------------
// Kernel 0: spectral normalization. sigma_max(A) via power iteration on A^T A
// (16x16). One wave; lanes 0-15 active. An = A * min(1, 0.99/sigma_max).
// ---------------------------------------------------------------------------
__global__ void ssm_spectral(const float* __restrict__ A, float* __restrict__ An) {
  __shared__ float lv[16], lu[16];
  const int lane = threadIdx.x;
  float Ar[16], At[16];
  if (lane < 16) {
    for (int k = 0; k < 16; ++k) {
      Ar[k] = A[lane * 16 + k];   // row lane of A
      At[k] = A[k * 16 + lane];   // column lane of A
    }
    lv[lane] = 0.25f;             // unit-norm start vector
  }
  __syncthreads();
  float sig2 = 1.0f;
  for (int it = 0; it < 64; ++it) {
    float u = 0.f;
    if (lane < 16) {
      for (int k = 0; k < 16; ++k) u = fmaf(Ar[k], lv[k], u);  // u = A v
      lu[lane] = u;
    }
    __syncthreads();
    float w = 0.f;
    if (lane < 16)
      for (int k = 0; k < 16; ++k) w = fmaf(At[k], lu[k], w);  // w = A^T u
    float n2 = w * w;
    n2 += __shfl_xor(n2, 8, 16); n2 += __shfl_xor(n2, 4, 16);
    n2 += __shfl_xor(n2, 2, 16); n2 += __shfl_xor(n2, 1, 16);
    float nrm = sqrtf(n2);                 // ||A^T A v|| -> sigma_max^2
    sig2 = nrm;
    if (lane < 16) lv[lane] = w / fmaxf(nrm, 1e-30f);
    __syncthreads();
  }
  float s0 = sqrtf(sig2);
  float scale = (s0 > 0.99f) ? (0.99f / s0) : 1.0f;
  if (lane < 16)
    for (int k = 0; k < 16; ++k) An[lane * 16 + k] = Ar[k] * scale;
}

// ---------------------------------------------------------------------------
// Kernel 1: Bx = x @ W_B^T (tokens x 16) and dt = clip(softplus(x.dt_w + b)).
// One wave per 16-token tile; 256 chained v_wmma_f32_16x16x4_f32 (K=1024).
// A operand (16x4 f32): lane m holds x[tok m, k0+koff .. +1], koff = (lane>=16)*2
// B operand (4x16 f32): lane n holds W_B[n, k0+koff .. +1]   (same addressing!)
// dt dot reuses the already-loaded x registers (fused, no extra x traffic).
// ---------------------------------------------------------------------------
__global__ void ssm_phase1(const float* __restrict__ x,
                           const float* __restrict__ W_B,
                           const float* __restrict__ dt_w,
                           const float* __restrict__ dt_b,
                           float* __restrict__ Bx, float* __restrict__ dtb) {
  const int lane = threadIdx.x & 31;
  const int wave = threadIdx.x >> 5;
  const int tile = blockIdx.x * (blockDim.x >> 5) + wave;
  if (tile >= NTILE) return;
  const int m    = lane & 15;
  const int koff = (lane >> 4) << 1;      // 0 (lanes 0-15) / 2 (lanes 16-31)
  const float* xrow = x    + (size_t)(tile * 16 + m) * DDIM + koff;
  const float* wrow = W_B  + (size_t)m * DDIM + koff;
  const float* dwp  = dt_w + koff;

  v8f c = {};
  float accdt = 0.f;
  for (int k0 = 0; k0 < DDIM; k0 += 4) {
    v2f a = *(const v2f*)(xrow + k0);
    v2f b = *(const v2f*)(wrow + k0);
    v2f w = *(const v2f*)(dwp + k0);
    accdt = fmaf(a.x, w.x, fmaf(a.y, w.y, accdt));
    c = wmma4(a, b, c);
  }

  // D layout: VGPR r = (M=r, N=lane) lanes 0-15 ; (M=r+8, N=lane-16) lanes 16-31
  const int mbase = (lane >> 4) << 3;
  float* bxp = Bx + (size_t)(tile * 16) * SDIM + m;
#pragma unroll
  for (int r = 0; r < 8; ++r)
    bxp[(size_t)(mbase + r) * SDIM] = c[r];

  // dt: lane m holds k≡{koff,koff+1} mod 4 partials; fold halves, softplus, clip
  float tot = accdt + __shfl_xor(accdt, 16, 32);
  if (lane < 16) {
    float z  = tot + dt_b[0];
    float sp = (z > 20.f) ? z : log1pf(__expf(z));
    sp = fminf(fmaxf(sp, 0.001f), 0.1f);
    dtb[tile * 16 + m] = sp;
  }
}

// ---------------------------------------------------------------------------
// Kernel 2: the sequential scan. 1 block, 64 threads = 2 waves.
// Lane layout: b = tid/16 (2 batches per wave), j = tid%16. State lives in a
// register; the 16x16 matvec uses 16 ds_bpermute broadcasts; LayerNorm mean/var
// use width-16 shfl_xor trees. No LDS, no barriers on the critical path.
// ---------------------------------------------------------------------------
__global__ void ssm_scan(const float* __restrict__ state0,
                         const float* __restrict__ An,
                         const float* __restrict__ dtb,
                         const float* __restrict__ Bx,
                         const float* __restrict__ ln_w,
                         const float* __restrict__ ln_b,
                         float* __restrict__ states,
                         float* __restrict__ newstate) {
  const int tid  = threadIdx.x;          // 0..63
  const int b    = tid >> 4;
  const int j    = tid & 15;
  const int half = tid & 16;             // base lane of my 16-lane group
  float Ar[16];
#pragma unroll
  for (int k = 0; k < 16; ++k) Ar[k] = An[j * 16 + k];
  const float lw = ln_w[j], lb = ln_b[j];
  float s = state0[b * SDIM + j];

  const float* dtp = dtb + (size_t)b * LDIM;
  const float* bxp = Bx + (size_t)b * LDIM * SDIM + j;
  float* stp = states + (size_t)b * LDIM * SDIM + j;

  for (int t = 0; t < LDIM; ++t) {
    float dt_t = dtp[t];                      // broadcast load
    float bx   = bxp[(size_t)t * SDIM];
    float acc = 0.f;
#pragma unroll
    for (int k = 0; k < 16; ++k) {
      float sk = __shfl(s, half + k, 32);     // ds_bpermute broadcast of s[k]
      acc = fmaf(Ar[k], sk, acc);
    }
    float sn = fmaf(dt_t, acc, s) + bx;
    // LayerNorm over the 16 state lanes
    float mu = sn;
    mu += __shfl_xor(mu, 8, 16); mu += __shfl_xor(mu, 4, 16);
    mu += __shfl_xor(mu, 2, 16); mu += __shfl_xor(mu, 1, 16);
    mu *= 0.0625f;
    float d  = sn - mu;
    float vv = d * d;
    vv += __shfl_xor(vv, 8, 16); vv += __shfl_xor(vv, 4, 16);
    vv += __shfl_xor(vv, 2, 16); vv += __shfl_xor(vv, 1, 16);
    vv *= 0.0625f;
    float out = fmaf(d * __frsqrt_rn(vv + 1e-5f), lw, lb);
    stp[(size_t)t * SDIM] = out;
    s = out;
  }
  newstate[b * SDIM + j] = s;
}

// ---------------------------------------------------------------------------
// Kernel 3: y = states @ W_C^T + D * x. One wave per 16-token tile; the tile's
// A operand (states, K=16) is loaded once into 4 v2f chunks, then 64 dim-tiles
// of 4 chained WMMAs each, with the D*x epilogue fused into the store.
// ---------------------------------------------------------------------------
__global__ void ssm_phase3(const float* __restrict__ states,
                           const float* __restrict__ W_C,
                           const float* __restrict__ Dv,
                           const float* __restrict__ x,
                           float* __restrict__ y) {
  const int lane = threadIdx.x & 31;
  const int wave = threadIdx.x >> 5;
  const int tile = blockIdx.x * (blockDim.x >> 5) + wave;
  if (tile >= NTILE) return;
  const int m     = lane & 15;
  const int koff  = (lane >> 4) << 1;
  const int mbase = (lane >> 4) << 3;

  const float* srow = states + (size_t)(tile * 16 + m) * SDIM + koff;
  v2f a0 = *(const v2f*)(srow + 0);
  v2f a1 = *(const v2f*)(srow + 4);
  v2f a2 = *(const v2f*)(srow + 8);
  v2f a3 = *(const v2f*)(srow + 12);

  for (int dt = 0; dt < DDIM / 16; ++dt) {
    const float* wc = W_C + (size_t)(dt * 16 + m) * SDIM + koff;
    v2f b0 = *(const v2f*)(wc + 0);
    v2f b1 = *(const v2f*)(wc + 4);
    v2f b2 = *(const v2f*)(wc + 8);
    v2f b3 = *(const v2f*)(wc + 12);
    v8f c = {};
    c = wmma4(a0, b0, c);
    c = wmma4(a1, b1, c);
    c = wmma4(a2, b2, c);
    c = wmma4(a3, b3, c);
    const float dd = Dv[dt * 16 + m];
#pragma unroll
    for (int r = 0; r < 8; ++r) {
      size_t row = (size_t)(tile * 16 + mbase + r);
      size_t idx = row * DDIM + (size_t)(dt * 16 + m);
      y[idx] = fmaf(dd, x[idx], c[r]);
    }
  }
}

// ---------------------------------------------------------------------------
extern "C" void kernel_launch(void* const* d_in, const int* in_sizes, int n_in,
                              void* d_out, int out_size, void* d_ws, size_t ws_size,
                              hipStream_t stream) {
  const float* x      = (const float*)d_in[0];
  const float* state0 = (const float*)d_in[1];
  const float* A      = (const float*)d_in[2];
  const float* W_B    = (const float*)d_in[3];
  const float* W_C    = (const float*)d_in[4];
  const float* Dv     = (const float*)d_in[5];
  const float* dt_w   = (const float*)d_in[6];
  const float* dt_b   = (const float*)d_in[7];
  const float* ln_w   = (const float*)d_in[8];
  const float* ln_b   = (const float*)d_in[9];

  float* y        = (float*)d_out;
  float* newstate = y + (size_t)NTOK * DDIM;     // tuple output #2

  // workspace layout (floats): An[256] | dt[16384] | Bx[262144] | states[262144]
  float* ws     = (float*)d_ws;
  float* An     = ws;
  float* dtb    = An + 256;
  float* Bx     = dtb + NTOK;
  float* states = Bx + (size_t)NTOK * SDIM;

  ssm_spectral<<<1, 32, 0, stream>>>(A, An);
  ssm_phase1<<<NTILE / 4, 128, 0, stream>>>(x, W_B, dt_w, dt_b, Bx, dtb);
  ssm_scan<<<1, 64, 0, stream>>>(state0, An, dtb, Bx, ln_w, ln_b, states, newstate);
  ssm_phase3<<<NTILE / 4, 128, 0, stream>>>(states, W_C, Dv, x, y);
}